// LocalWLGNN_5892695130447
// MI455X (gfx1250) — compile-verified
//
#include <hip/hip_runtime.h>

typedef __attribute__((ext_vector_type(2))) float v2f;
typedef __attribute__((ext_vector_type(8))) float v8f;

// ---------------------------------------------------------------------------
// h = x  (row-wise float4 copy)
// ---------------------------------------------------------------------------
__global__ __launch_bounds__(256) void copy_f4_kernel(const float4* __restrict__ s,
                                                      float4* __restrict__ d, int n4) {
  int i = blockIdx.x * blockDim.x + threadIdx.x;
  if (i < n4) d[i] = s[i];
}

// ---------------------------------------------------------------------------
// h[dst[e]][:] += x[src[e]][:]   one thread per (edge, feature)
// lane 0..31 covers 32 consecutive features of one edge -> coalesced 128B.
// Hardware f32 atomic via inline CDNA5 asm (avoids CAS-loop fallback).
// ---------------------------------------------------------------------------
__global__ __launch_bounds__(256) void scatter_add_kernel(const float* __restrict__ x,
                                                          float* __restrict__ h,
                                                          const int* __restrict__ src,
                                                          const int* __restrict__ dst,
                                                          int E) {
  long long tid = (long long)blockIdx.x * blockDim.x + threadIdx.x;
  int e = (int)(tid >> 6);
  int f = (int)(tid & 63);
  if (e < E) {
    int s = src[e];
    int t = dst[e];
    float v = x[(long long)s * 64 + f];
    float* p = &h[(long long)t * 64 + f];
    asm volatile("global_atomic_add_f32 %0, %1, off" : : "v"(p), "v"(v) : "memory");
  }
}

// ---------------------------------------------------------------------------
// Out[16 rows x 64 cols] (+)= A_tile(16x64) @ W(64x64) using V_WMMA_F32_16X16X4_F32.
// Block = 128 threads = 4 waves; wave w owns 16x16 output tile at colBase=16*w.
// ACCUM=0: C initialized to (1+eps[layer]) * Xin   (fused layer-output init)
// ACCUM=1: C initialized from Out (accumulate second hop)
// ---------------------------------------------------------------------------
template <int ACCUM>
__global__ __launch_bounds__(128) void gemm64_wmma(const float* __restrict__ A,
                                                   const float* __restrict__ W,
                                                   const float* __restrict__ Xin,
                                                   float* __restrict__ Out,
                                                   const float* __restrict__ eps,
                                                   int layer, int nrows) {
  __shared__ float sW[64 * 64];      // full 64x64 weight, row-major
  __shared__ float sA[16 * 66];      // 16x64 A tile, pitch 66 (bank-conflict pad)

  const int tid = threadIdx.x;
  const int rowBase = blockIdx.x * 16;

  // Stage W: 4096 floats / 128 threads = 8 float4 each.
  {
    const float4* Wv = (const float4*)W;
    float4* sWv = (float4*)sW;
#pragma unroll
    for (int i = 0; i < 8; ++i) sWv[tid + i * 128] = Wv[tid + i * 128];
  }
  // Stage A tile: 1024 floats / 128 threads = 8 each (coalesced global read).
#pragma unroll
  for (int i = 0; i < 8; ++i) {
    int idx = tid + i * 128;                 // 0..1023
    int r = idx >> 6, c = idx & 63;
    int row = rowBase + r;
    if (row >= nrows) row = nrows - 1;       // clamp (keeps EXEC full)
    sA[r * 66 + c] = A[(long long)row * 64 + c];
  }
  __syncthreads();

  const int lane = tid & 31;
  const int wave = tid >> 5;
  const int hi = lane >> 4;                  // lane half: K-pair select
  const int m = lane & 15;                   // row (A) / col (B,C) within tile
  const int colBase = wave * 16;
  const int col = colBase + m;

  // C init (fused): either accumulate existing Out, or (1+eps)*Xin.
  v8f c;
  if (ACCUM) {
#pragma unroll
    for (int r = 0; r < 8; ++r) {
      int row = rowBase + r + hi * 8;
      if (row >= nrows) row = nrows - 1;
      c[r] = Out[(long long)row * 64 + col];
    }
  } else {
    const float alpha = 1.0f + eps[layer];
#pragma unroll
    for (int r = 0; r < 8; ++r) {
      int row = rowBase + r + hi * 8;
      if (row >= nrows) row = nrows - 1;
      c[r] = alpha * Xin[(long long)row * 64 + col];
    }
  }

  // 16 k-steps of 16x16x4 f32 WMMA (K = 64).
#pragma unroll
  for (int ks = 0; ks < 16; ++ks) {
    const int kb = ks * 4 + hi * 2;          // this lane-half's K pair
    v2f a = *(const v2f*)&sA[m * 66 + kb];   // A[m][kb], A[m][kb+1]
    v2f b;
    b.x = sW[(kb + 0) * 64 + col];           // B[kb][col]
    b.y = sW[(kb + 1) * 64 + col];           // B[kb+1][col]
    c = __builtin_amdgcn_wmma_f32_16x16x4_f32(false, a, false, b, (short)0, c,
                                              false, false);
  }

#pragma unroll
  for (int r = 0; r < 8; ++r) {
    int row = rowBase + r + hi * 8;
    if (row < nrows) Out[(long long)row * 64 + col] = c[r];
  }
}

// ---------------------------------------------------------------------------
// Head: Out[N x 40] = X[N x 64] @ Wh[64 x 40]. 3 col-tiles (last zero-padded),
// block = 96 threads = 3 waves, one 16x16 tile per wave, masked store col<40.
// ---------------------------------------------------------------------------
__global__ __launch_bounds__(96) void head_wmma(const float* __restrict__ X,
                                                const float* __restrict__ Wh,
                                                float* __restrict__ Out, int nrows) {
  __shared__ float sW[64 * 48];              // 64x40 padded to 64x48
  const int tid = threadIdx.x;
  for (int i = tid; i < 64 * 48; i += 96) sW[i] = 0.0f;
  __syncthreads();
  for (int i = tid; i < 64 * 40; i += 96) {
    int r = i / 40, cc = i - r * 40;
    sW[r * 48 + cc] = Wh[i];
  }
  __syncthreads();

  const int lane = tid & 31;
  const int wave = tid >> 5;                 // 0..2
  const int hi = lane >> 4;
  const int m = lane & 15;
  const int colBase = wave * 16;
  const int rowBase = blockIdx.x * 16;

  v8f c = {};
#pragma unroll
  for (int ks = 0; ks < 16; ++ks) {
    const int kb = ks * 4 + hi * 2;
    int row = rowBase + m;
    if (row >= nrows) row = nrows - 1;
    v2f a = *(const v2f*)&X[(long long)row * 64 + kb];
    v2f b;
    b.x = sW[(kb + 0) * 48 + colBase + m];
    b.y = sW[(kb + 1) * 48 + colBase + m];
    c = __builtin_amdgcn_wmma_f32_16x16x4_f32(false, a, false, b, (short)0, c,
                                              false, false);
  }

  const int col = colBase + m;
#pragma unroll
  for (int r = 0; r < 8; ++r) {
    int row = rowBase + r + hi * 8;
    if (row < nrows && col < 40) Out[(long long)row * 40 + col] = c[r];
  }
}

// ---------------------------------------------------------------------------
extern "C" void kernel_launch(void* const* d_in, const int* in_sizes, int n_in,
                              void* d_out, int out_size, void* d_ws, size_t ws_size,
                              hipStream_t stream) {
  const float* x     = (const float*)d_in[0];
  const float* W00   = (const float*)d_in[1];
  const float* W01   = (const float*)d_in[2];
  const float* W10   = (const float*)d_in[3];
  const float* W11   = (const float*)d_in[4];
  const float* Whead = (const float*)d_in[5];
  const float* eps   = (const float*)d_in[6];
  const int* src0    = (const int*)d_in[7];
  const int* dst0    = (const int*)d_in[8];
  const int* src1    = (const int*)d_in[9];
  const int* dst1    = (const int*)d_in[10];

  const int N = in_sizes[0] / 64;
  const int E = in_sizes[7];

  float* buf0 = (float*)d_ws;                   // layer-0 output
  float* buf1 = buf0 + (size_t)N * 64;          // layer-1 output
  float* hbuf = buf1 + (size_t)N * 64;          // aggregation scratch

  const int n4 = N * 16;                        // float4 count of an N x 64 buffer
  const dim3 copyGrid((n4 + 255) / 256), copyBlk(256);
  const long long scatterThreads = (long long)E * 64;
  const dim3 scatGrid((unsigned)((scatterThreads + 255) / 256)), scatBlk(256);
  const dim3 gemmGrid((N + 15) / 16), gemmBlk(128);
  const dim3 headBlk(96);

  const float* Wl[2][2] = {{W00, W01}, {W10, W11}};
  const int* srcs[2] = {src0, src1};
  const int* dsts[2] = {dst0, dst1};

  const float* xin = x;
  float* bufs[2] = {buf0, buf1};

  for (int l = 0; l < 2; ++l) {
    float* out = bufs[l];
    // hop 0: h = scatter_agg(xin);  out = (1+eps[l])*xin + h @ W[l][0]
    copy_f4_kernel<<<copyGrid, copyBlk, 0, stream>>>((const float4*)xin,
                                                     (float4*)hbuf, n4);
    scatter_add_kernel<<<scatGrid, scatBlk, 0, stream>>>(xin, hbuf, srcs[0],
                                                         dsts[0], E);
    gemm64_wmma<0><<<gemmGrid, gemmBlk, 0, stream>>>(hbuf, Wl[l][0], xin, out,
                                                     eps, l, N);
    // hop 1: h = scatter_agg(xin);  out += h @ W[l][1]
    copy_f4_kernel<<<copyGrid, copyBlk, 0, stream>>>((const float4*)xin,
                                                     (float4*)hbuf, n4);
    scatter_add_kernel<<<scatGrid, scatBlk, 0, stream>>>(xin, hbuf, srcs[1],
                                                         dsts[1], E);
    gemm64_wmma<1><<<gemmGrid, gemmBlk, 0, stream>>>(hbuf, Wl[l][1], nullptr, out,
                                                     eps, l, N);
    xin = out;
  }

  head_wmma<<<gemmGrid, headBlk, 0, stream>>>(xin, Whead, (float*)d_out, N);
}